// GraphSAGE_6064493822170
// MI455X (gfx1250) — compile-verified
//
#include <hip/hip_runtime.h>
#include <hip/hip_bf16.h>

typedef __attribute__((ext_vector_type(2))) float v2f;
typedef __attribute__((ext_vector_type(8))) float v8f;

// XOR-swizzled LDS addressing: keeps pitch 64 (64KB total LDS) while making
// WMMA fragment loads (fixed col offset, varying row across lanes) hit
// distinct banks: bank = (col ^ 4*(row&15)) % 64 -> conflict-free b64 loads.
// Mask is a multiple of 4, so 2-float (b64) and 4-float (b128) groups whose
// base is 2-/4-aligned stay contiguous and aligned.
__device__ __forceinline__ int swz(int row, int col) {
    return row * 64 + (col ^ ((row & 15) << 2));
}

// ---------------------------------------------------------------------------
// Zero fill
// ---------------------------------------------------------------------------
__global__ void zero_kernel(float* __restrict__ p, long long n) {
    long long i = (long long)blockIdx.x * blockDim.x + threadIdx.x;
    if (i < n) p[i] = 0.0f;
}

// ---------------------------------------------------------------------------
// Edge scatter: agg[dst] += feat[src] (64 channels), optional degree count.
// 16 threads per edge, 4 channels each (float4 gather + 4 f32 atomics).
// agg fits in L2 (25.6MB of 192MB) -> atomics stay on-chip.
// ---------------------------------------------------------------------------
__global__ void scatter_kernel(const float* __restrict__ feat,
                               const int* __restrict__ src,
                               const int* __restrict__ dst,
                               float* __restrict__ agg,
                               float* __restrict__ cnt,
                               int n_edges, int do_cnt) {
    long long tid = (long long)blockIdx.x * blockDim.x + threadIdx.x;
    int e = (int)(tid >> 4);
    if (e >= n_edges) return;
    int c = ((int)tid & 15) * 4;
    int s = src[e];
    int d = dst[e];
    const float4 v = *(const float4*)(feat + (long long)s * 64 + c);
    float* o = agg + (long long)d * 64 + c;
    atomicAdd(o + 0, v.x);
    atomicAdd(o + 1, v.y);
    atomicAdd(o + 2, v.z);
    atomicAdd(o + 3, v.w);
    if (do_cnt && ((tid & 15) == 0)) atomicAdd(cnt + d, 1.0f);
}

// ---------------------------------------------------------------------------
// Fused SAGEConv GEMM: out = relu( (agg/max(cnt,1)) @ Wagg + root @ Wroot + b )
// 128 threads = 4 waves; 64 nodes per block; 16-node tile per wave.
// V_WMMA_F32_16X16X4_F32 (full fp32 matrix pipe); A and B fragments are
// single ds_load_b64 from swizzled LDS (weights stored transposed).
// ---------------------------------------------------------------------------
__global__ void __launch_bounds__(128)
sage_gemm_kernel(const float* __restrict__ aggin,   // [N,64] raw sums
                 const float* __restrict__ cnt,     // [N]
                 const float* __restrict__ rootin,  // [N,64]
                 const float* __restrict__ Wagg,    // [64,64] row-major (k, outcol)
                 const float* __restrict__ Wroot,   // [64,64] row-major
                 const float* __restrict__ bias,    // [64]
                 float* __restrict__ out,           // [N,64]
                 int n_nodes) {
    __shared__ __attribute__((aligned(16))) float sWa[64 * 64]; // transposed: [outcol][k]
    __shared__ __attribute__((aligned(16))) float sWr[64 * 64]; // transposed: [outcol][k]
    __shared__ __attribute__((aligned(16))) float sA[64 * 64];  // [node][k] mean-agg
    __shared__ __attribute__((aligned(16))) float sR[64 * 64];  // [node][k] root

    const int t = threadIdx.x;
    const int nodeBase = blockIdx.x * 64;

    // Stage weights transposed into swizzled LDS (coalesced b128 global reads).
    for (int i = t * 4; i < 4096; i += 512) {
        const int row = i >> 6;   // k index
        const int c   = i & 63;   // out-col base
        const float4 wa = *(const float4*)(Wagg + i);
        const float4 wr = *(const float4*)(Wroot + i);
        sWa[swz(c + 0, row)] = wa.x;
        sWa[swz(c + 1, row)] = wa.y;
        sWa[swz(c + 2, row)] = wa.z;
        sWa[swz(c + 3, row)] = wa.w;
        sWr[swz(c + 0, row)] = wr.x;
        sWr[swz(c + 1, row)] = wr.y;
        sWr[swz(c + 2, row)] = wr.z;
        sWr[swz(c + 3, row)] = wr.w;
    }
    // Stage activations (swizzled, b128 stores); fuse mean division.
    for (int i = t * 4; i < 4096; i += 512) {
        const int rl = i >> 6;        // local node row
        const int cc = i & 63;        // k base (multiple of 4)
        const int row = nodeBase + rl;
        float4 a, r;
        if (row < n_nodes) {
            float c = cnt[row];
            c = (c < 1.0f) ? 1.0f : c;
            const float inv = 1.0f / c;
            a = *(const float4*)(aggin + (long long)row * 64 + cc);
            a.x *= inv; a.y *= inv; a.z *= inv; a.w *= inv;
            r = *(const float4*)(rootin + (long long)row * 64 + cc);
        } else {
            a = make_float4(0.f, 0.f, 0.f, 0.f);
            r = a;
        }
        *(float4*)(&sA[swz(rl, cc)]) = a;
        *(float4*)(&sR[swz(rl, cc)]) = r;
    }
    __syncthreads();

    const int wave = t >> 5;
    const int lane = t & 31;
    const int tile = blockIdx.x * 4 + wave;     // 16-node tile index
    if (tile * 16 >= n_nodes) return;           // wave-uniform -> EXEC stays all-1s

    const int mrow  = lane & 15;                // A row in tile / B,C column
    const int khalf = (lane >> 4) * 2;          // ISA 32b A layout: K {0,1} vs {2,3}
    const int arow  = wave * 16 + mrow;

    v8f acc[4];
    acc[0] = v8f{}; acc[1] = v8f{}; acc[2] = v8f{}; acc[3] = v8f{};

#pragma unroll
    for (int k = 0; k < 16; ++k) {
        const int kb = k * 4 + khalf;           // even -> 8B-aligned after swizzle
        const v2f aA = *(const v2f*)(&sA[swz(arow, kb)]);
        const v2f aR = *(const v2f*)(&sR[swz(arow, kb)]);
#pragma unroll
        for (int j = 0; j < 4; ++j) {
            const int col = j * 16 + mrow;
            const v2f bA = *(const v2f*)(&sWa[swz(col, kb)]);
            const v2f bR = *(const v2f*)(&sWr[swz(col, kb)]);
            acc[j] = __builtin_amdgcn_wmma_f32_16x16x4_f32(
                false, aA, false, bA, (short)0, acc[j], false, false);
            acc[j] = __builtin_amdgcn_wmma_f32_16x16x4_f32(
                false, aR, false, bR, (short)0, acc[j], false, false);
        }
    }

    // C/D layout: VGPR r -> M=r (lanes 0-15), M=r+8 (lanes 16-31); N = lane&15.
    const int rowHi = (lane >> 4) * 8;
#pragma unroll
    for (int j = 0; j < 4; ++j) {
        const int col = j * 16 + mrow;
        const float bb = bias[col];
#pragma unroll
        for (int r = 0; r < 8; ++r) {
            const int grow = tile * 16 + rowHi + r;
            float v = acc[j][r] + bb;
            out[(long long)grow * 64 + col] = (v > 0.0f) ? v : 0.0f;
        }
    }
}

// ---------------------------------------------------------------------------
// Output head: out[n, 0:2] = h2[n,:] @ W_lin + b_lin   (64x2, memory bound)
// ---------------------------------------------------------------------------
__global__ void lin_kernel(const float* __restrict__ h,
                           const float* __restrict__ W,   // [64,2] row-major
                           const float* __restrict__ b,   // [2]
                           float* __restrict__ out, int n) {
    int i = blockIdx.x * blockDim.x + threadIdx.x;
    if (i >= n) return;
    const float* row = h + (long long)i * 64;
    float o0 = b[0], o1 = b[1];
#pragma unroll 4
    for (int c = 0; c < 64; ++c) {
        float v = row[c];
        o0 += v * W[c * 2 + 0];
        o1 += v * W[c * 2 + 1];
    }
    out[(long long)i * 2 + 0] = o0;
    out[(long long)i * 2 + 1] = o1;
}

// ---------------------------------------------------------------------------
// Launch
// ---------------------------------------------------------------------------
extern "C" void kernel_launch(void* const* d_in, const int* in_sizes, int n_in,
                              void* d_out, int out_size, void* d_ws, size_t ws_size,
                              hipStream_t stream) {
    const float* x    = (const float*)d_in[0];
    const int*   ei   = (const int*)d_in[1];
    const float* W1a  = (const float*)d_in[2];
    const float* W1r  = (const float*)d_in[3];
    const float* b1   = (const float*)d_in[4];
    const float* W2a  = (const float*)d_in[5];
    const float* W2r  = (const float*)d_in[6];
    const float* b2   = (const float*)d_in[7];
    const float* Wl   = (const float*)d_in[8];
    const float* bl   = (const float*)d_in[9];

    const int Nn = in_sizes[0] / 64;   // 100000
    const int E  = in_sizes[1] / 2;    // 1600000
    const int* src = ei;
    const int* dst = ei + E;

    // Workspace: agg [N*64] | cnt [N] | h1 [N*64] | h2 [N*64]  (~77 MB)
    float* agg = (float*)d_ws;
    float* cnt = agg + (size_t)Nn * 64;
    float* h1  = cnt + Nn;
    float* h2  = h1 + (size_t)Nn * 64;

    const long long scatterThreads = (long long)E * 16;
    const int scatterBlocks = (int)((scatterThreads + 255) / 256);
    const int gemmBlocks = (Nn + 63) / 64;

    // ---- Layer 1 ----
    {
        long long z = (long long)Nn * 64 + Nn;   // agg + cnt contiguous
        zero_kernel<<<(int)((z + 255) / 256), 256, 0, stream>>>(agg, z);
    }
    scatter_kernel<<<scatterBlocks, 256, 0, stream>>>(x, src, dst, agg, cnt, E, 1);
    sage_gemm_kernel<<<gemmBlocks, 128, 0, stream>>>(agg, cnt, x, W1a, W1r, b1, h1, Nn);

    // ---- Layer 2 (counts reused; depend only on dst) ----
    {
        long long z = (long long)Nn * 64;
        zero_kernel<<<(int)((z + 255) / 256), 256, 0, stream>>>(agg, z);
    }
    scatter_kernel<<<scatterBlocks, 256, 0, stream>>>(h1, src, dst, agg, cnt, E, 0);
    sage_gemm_kernel<<<gemmBlocks, 128, 0, stream>>>(agg, cnt, h1, W2a, W2r, b2, h2, Nn);

    // ---- Output head ----
    lin_kernel<<<(Nn + 255) / 256, 256, 0, stream>>>(h2, Wl, bl, (float*)d_out, Nn);
}